// preDeepMD_52888227283576
// MI455X (gfx1250) — compile-verified
//
#include <hip/hip_runtime.h>

// Problem constants (from reference)
#define BB   4
#define NN   2048
#define AA   128       // neighbors, also block size of kernel 1
#define E1   25
#define E2   50
#define E3   100       // M
#define F0   1600      // 16*M
#define F1   240
#define KC   64        // K-chunk staged per TDM transfer in the fitting net
#define RMIN_F 5.8f
#define RMAX_F 6.0f
#define PI_F 3.14159265358979323846f

typedef __attribute__((ext_vector_type(2))) float v2f;
typedef __attribute__((ext_vector_type(8))) float v8f;
typedef __attribute__((ext_vector_type(4))) unsigned int u32x4;
typedef __attribute__((ext_vector_type(8))) unsigned int u32x8;
typedef __attribute__((ext_vector_type(4))) int i32x4;

__device__ __forceinline__ float fast_tanh(float x) {
    // tanh(x) = (e^{2x}-1)/(e^{2x}+1), e^x via hardware v_exp_f32
    float e = __expf(2.0f * x);
    return (e - 1.0f) / (e + 1.0f);
}

// Low 32 bits of a generic pointer to LDS == byte offset in the wave's LDS
// allocation (aperture bits live in [63:32] per the CDNA5 aperture check).
__device__ __forceinline__ unsigned int lds_off(const void* p) {
    return (unsigned int)(unsigned long long)p;
}

// -------------------------------------------------------------------------
// Tensor Data Mover: contiguous copy of `ndw` dwords global -> LDS.
// D# built per ISA 8.3/8.4: group0 = {count=1, lds_addr, global_addr, type=2},
// group1 = {data_size=4B, tensor_dim0=ndw, tensor_dim1=1, tile_dim0=ndw,
//           tile_dim1=1, tensor_dim0_stride=ndw}; groups 2/3 zero (<=2D).
// Issue from ONE wave only; tracked with TENSORcnt.
// -------------------------------------------------------------------------
__device__ __forceinline__ void tdm_load_dwords(unsigned int lds_byte,
                                                const void* gsrc,
                                                unsigned int ndw)
{
    unsigned long long ga = (unsigned long long)gsrc;
    u32x4 g0;
    g0[0] = 1u;                                        // count=1, user mode
    g0[1] = lds_byte;                                  // LDS dest (bytes)
    g0[2] = (unsigned int)ga;                          // global_addr[31:0]
    g0[3] = (unsigned int)((ga >> 32) & 0x1FFFFFFu)    // global_addr[56:32]
          | (2u << 30);                                // type = 2 ("image")
    u32x8 g1;
    g1[0] = 2u << 16;                                  // data_size = 4 bytes
    g1[1] = (ndw & 0xFFFFu) << 16;                     // tensor_dim0 lo16
    g1[2] = ((ndw >> 16) & 0xFFFFu) | (1u << 16);      // tensor_dim0 hi16, tensor_dim1=1
    g1[3] = (ndw & 0xFFFFu) << 16;                     // tile_dim0 (<=65535)
    g1[4] = 1u;                                        // tile_dim1 = 1
    g1[5] = ndw;                                       // tensor_dim0_stride lo32
    g1[6] = 0u;
    g1[7] = 0u;
    i32x4 z4 = {0, 0, 0, 0};
    asm volatile("tensor_load_to_lds %0, %1, %2, %3"
                 :: "s"(g0), "s"(g1), "s"(z4), "s"(z4)
                 : "memory");
}

// -------------------------------------------------------------------------
// Kernel 1: one workgroup per atom (b,n). 128 threads = 1 neighbor each.
// Geometry -> S_Rij/Ri -> embedding MLP (G stays in LDS, never HBM)
// -> tmpA/tmpB -> DR via V_WMMA_F32_16X16X4_F32 (K=4 exact, 1 wmma/tile).
// Big weight tables staged by TDM, overlapped with the geometry phase.
// -------------------------------------------------------------------------
__global__ __launch_bounds__(AA) void k_embed_dr(
    const float* __restrict__ image_dR, const int* __restrict__ list_neigh,
    const float* __restrict__ We0, const float* __restrict__ be0,
    const float* __restrict__ We1, const float* __restrict__ be1,
    const float* __restrict__ We2, const float* __restrict__ be2,
    float* __restrict__ dr_out)
{
    __shared__ float sW0[E1], sB0[E1];
    __shared__ float sW1[E1 * E2], sB1[E2];
    __shared__ float sW2[E2 * E3], sB2[E3];
    __shared__ float sRi[AA][4];
    __shared__ float sG[AA][E3 + 1];      // stride 101 -> conflict-free columns
    __shared__ int   sMask[AA];
    __shared__ float sTmpA[16][4];
    __shared__ float sTmpB[4][112];       // N padded 100 -> 112 (7 tiles of 16)

    const int tid = threadIdx.x;          // == neighbor index a
    const int bn  = blockIdx.x;           // atom index in [0, B*N)
    const int lane = tid & 31, wid = tid >> 5;

    // ---- async-stage the two large weight tables via the TDM ----
    if (wid == 0) {
        tdm_load_dwords(lds_off(sW1), We1, E1 * E2);
        tdm_load_dwords(lds_off(sW2), We2, E2 * E3);
    }
    // small tables loaded cooperatively
    for (int i = tid; i < E1; i += AA) { sW0[i] = We0[i]; sB0[i] = be0[i]; }
    for (int i = tid; i < E2; i += AA) sB1[i] = be1[i];
    for (int i = tid; i < E3; i += AA) sB2[i] = be2[i];

    // ---- geometry / smooth() (uses SQUARED distance, per reference) ----
    const int a = tid;
    const long gbase = ((long)bn * AA + a) * 3;
    const float dx = image_dR[gbase + 0];
    const float dy = image_dR[gbase + 1];
    const float dz = image_dR[gbase + 2];
    const int   nb = list_neigh[(long)bn * AA + a];
    const float r2 = dx * dx + dy * dy + dz * dz;
    const bool  mk    = nb > 0;
    const bool  mkmin = r2 < RMIN_F;
    const float safe  = (r2 > 0.0f) ? r2 : 1.0f;
    const float sinv  = 1.0f / safe;
    const float scos  = 0.5f * __cosf(PI_F * (r2 - RMIN_F) / (RMAX_F - RMIN_F)) + 0.5f;
    const float S = (mk && mkmin) ? sinv : ((!mkmin && r2 < RMAX_F) ? scos : 0.0f);

    sMask[a] = mk ? 1 : 0;
    if (wid == 0) __builtin_amdgcn_s_wait_tensorcnt(0);   // TDM done before barrier
    __syncthreads();   // covers weights (TDM + manual) and sMask

    // prefix = cumulative AND of mask (python loop breaks at first invalid)
    int pfx = 1;
    for (int j = 0; j <= a; ++j) pfx &= sMask[j];
    const float f = pfx ? S * sinv : 0.0f;
    sRi[a][0] = S;
    sRi[a][1] = f * dx;
    sRi[a][2] = f * dy;
    sRi[a][3] = f * dz;

    // ---- embedding MLP 1 -> 25 -> 50 -> 100 (per-point, VALU) ----
    float h1[E1];
    #pragma unroll
    for (int j = 0; j < E1; ++j) h1[j] = fast_tanh(S * sW0[j] + sB0[j]);

    float h2[E2];
    for (int k = 0; k < E2; ++k) {
        float acc = sB1[k];
        #pragma unroll
        for (int j = 0; j < E1; ++j) acc += h1[j] * sW1[j * E2 + k];
        h2[k] = fast_tanh(acc);
    }
    for (int l = 0; l < E3; ++l) {
        float acc = sB2[l];
        #pragma unroll
        for (int k = 0; k < E2; ++k) acc += h2[k] * sW2[k * E3 + l];
        sG[a][l] = fast_tanh(acc);
    }
    __syncthreads();

    // ---- tmpA[m][c] = sum_a G[a][m] * Ri[a][c], m<16 ----
    if (tid < 64) {
        const int m = tid >> 2, c = tid & 3;
        float acc = 0.0f;
        for (int q = 0; q < AA; ++q) acc += sG[q][m] * sRi[q][c];
        sTmpA[m][c] = acc;
    }
    // ---- tmpB[c][col] = sum_a Ri[a][c] * G[a][col], padded to 112 cols ----
    for (int idx = tid; idx < 4 * 112; idx += AA) {
        const int c = idx / 112, col = idx % 112;
        float acc = 0.0f;
        if (col < E3) {
            for (int q = 0; q < AA; ++q) acc += sRi[q][c] * sG[q][col];
        }
        sTmpB[c][col] = acc;
    }
    __syncthreads();

    // ---- DR(16x100) = tmpA(16x4) @ tmpB(4x112): one wmma per 16-col tile ----
    const int half = lane >> 4, l16 = lane & 15;
    for (int t = wid; t < 7; t += 4) {          // uniform per wave -> EXEC all 1s
        v2f afrag, bfrag;
        afrag.x = sTmpA[l16][2 * half + 0];     // A 16x4: lanes 0-15 K0/K1, 16-31 K2/K3
        afrag.y = sTmpA[l16][2 * half + 1];
        bfrag.x = sTmpB[2 * half + 0][t * 16 + l16];  // B 4x16: N on lanes
        bfrag.y = sTmpB[2 * half + 1][t * 16 + l16];
        v8f d = {0.f, 0.f, 0.f, 0.f, 0.f, 0.f, 0.f, 0.f};
        d = __builtin_amdgcn_wmma_f32_16x16x4_f32(false, afrag, false, bfrag,
                                                  (short)0, d, false, false);
        const int n = t * 16 + l16;
        if (n < E3) {
            float* drp = dr_out + (long)bn * F0 + n;
            #pragma unroll
            for (int v = 0; v < 8; ++v) drp[(v + 8 * half) * E3] = d[v];
        }
    }
}

// -------------------------------------------------------------------------
// Zero per-batch Etot accumulators (runs every launch -> deterministic).
// -------------------------------------------------------------------------
__global__ void k_zero_etot(float* __restrict__ out)
{
    if (threadIdx.x < BB) out[threadIdx.x] = 0.0f;
}

// -------------------------------------------------------------------------
// One fitting layer: [16 x K] (LDS) @ W[K x 240] (global, TDM-chunked into
// LDS) + bias, tanh -> Hout (LDS). 8 waves; wave w owns N-tiles {w, w+8};
// both accumulators live across K-chunks so the A fragment is loaded once.
// -------------------------------------------------------------------------
__device__ __forceinline__ void fit_layer(
    const float* __restrict__ Ain, int astride,   // LDS, [16][astride]
    float* __restrict__ Hout, int hstride,        // LDS, [16][hstride]
    float* __restrict__ sW,                       // LDS chunk buffer [KC*F1]
    const float* __restrict__ Wglob,              // global weights [K][F1]
    const float* __restrict__ bvec, int K, int tid)
{
    const int lane = tid & 31, wid = tid >> 5;
    const int half = lane >> 4, l16 = lane & 15;
    const int t0 = wid, t1 = wid + 8;
    const int n0 = t0 * 16 + l16;
    const int n1 = t1 * 16 + l16;
    const bool has2 = (t1 < 15);                  // wave-uniform

    v8f acc0 = {}, acc1 = {};
    for (int kc = 0; kc < K; kc += KC) {
        const int klen = (K - kc < KC) ? (K - kc) : KC;
        __syncthreads();                          // sW consumed by all waves
        if (wid == 0) {
            tdm_load_dwords(lds_off(sW), Wglob + (long)kc * F1,
                            (unsigned int)(klen * F1));
            __builtin_amdgcn_s_wait_tensorcnt(0);
        }
        __syncthreads();                          // sW chunk visible
        #pragma unroll 4
        for (int kk = 0; kk < klen; kk += 4) {
            v2f afrag;
            afrag.x = Ain[l16 * astride + kc + kk + 2 * half + 0];
            afrag.y = Ain[l16 * astride + kc + kk + 2 * half + 1];
            v2f b0;
            b0.x = sW[(kk + 2 * half + 0) * F1 + n0];
            b0.y = sW[(kk + 2 * half + 1) * F1 + n0];
            acc0 = __builtin_amdgcn_wmma_f32_16x16x4_f32(false, afrag, false, b0,
                                                         (short)0, acc0, false, false);
            if (has2) {
                v2f b1;
                b1.x = sW[(kk + 2 * half + 0) * F1 + n1];
                b1.y = sW[(kk + 2 * half + 1) * F1 + n1];
                acc1 = __builtin_amdgcn_wmma_f32_16x16x4_f32(false, afrag, false, b1,
                                                             (short)0, acc1, false, false);
            }
        }
    }
    // epilogue: bias + tanh (Hout never aliases Ain)
    const float b0s = bvec[n0];
    #pragma unroll
    for (int v = 0; v < 8; ++v)
        Hout[(v + 8 * half) * hstride + n0] = fast_tanh(acc0[v] + b0s);
    if (has2) {
        const float b1s = bvec[n1];
        #pragma unroll
        for (int v = 0; v < 8; ++v)
            Hout[(v + 8 * half) * hstride + n1] = fast_tanh(acc1[v] + b1s);
    }
}

// -------------------------------------------------------------------------
// Kernel 2: fused fitting net. One workgroup = 16 atom rows, 256 threads
// (8 waves). X tile staged in LDS; weights TDM-chunked into LDS.
// LDS ~196KB of the 320KB WGP pool (occupancy traded for on-chip reuse).
// -------------------------------------------------------------------------
__global__ __launch_bounds__(256) void k_fit(
    const float* __restrict__ dr_in,
    const float* __restrict__ Wf0, const float* __restrict__ bf0,
    const float* __restrict__ Wf1, const float* __restrict__ bf1,
    const float* __restrict__ Wf2, const float* __restrict__ bf2,
    const float* __restrict__ Wf3, const float* __restrict__ bf3,
    float* __restrict__ out)
{
    __shared__ float sX[16][F0 + 4];      // stride 1604 (even, mod 64 = 4)
    __shared__ float sWc[KC * F1];        // TDM-staged weight chunk (61.4KB)
    __shared__ float sHa[16][F1 + 2];     // stride 242 (even, conflict-free)
    __shared__ float sHb[16][F1 + 2];
    __shared__ float sW3[F1];
    __shared__ float sBf3;

    const int tid = threadIdx.x;
    const int wg  = blockIdx.x;           // row-block of 16 atoms

    // ---- stage X = DR rows [16 x 1600] into LDS with float4 loads ----
    const float4* src = (const float4*)(dr_in + (long)wg * 16 * F0);
    for (int i = tid; i < 16 * (F0 / 4); i += 256) {
        const int r = i / (F0 / 4), k4 = i % (F0 / 4);
        const float4 v = src[r * (F0 / 4) + k4];
        sX[r][k4 * 4 + 0] = v.x;
        sX[r][k4 * 4 + 1] = v.y;
        sX[r][k4 * 4 + 2] = v.z;
        sX[r][k4 * 4 + 3] = v.w;
    }
    if (tid < F1) sW3[tid] = Wf3[tid];
    if (tid == 0) sBf3 = bf3[0];
    __syncthreads();

    // ---- fitting layers 1600 -> 240 -> 240 -> 240 (tanh) ----
    fit_layer(&sX[0][0],  F0 + 4, &sHa[0][0], F1 + 2, sWc, Wf0, bf0, F0, tid);
    __syncthreads();
    fit_layer(&sHa[0][0], F1 + 2, &sHb[0][0], F1 + 2, sWc, Wf1, bf1, F1, tid);
    __syncthreads();
    fit_layer(&sHb[0][0], F1 + 2, &sHa[0][0], F1 + 2, sWc, Wf2, bf2, F1, tid);
    __syncthreads();

    // ---- layer 3: 240 -> 1 (no tanh), Ei + per-batch Etot ----
    if (tid < 16) {
        float acc = sBf3;
        for (int k = 0; k < F1; ++k) acc += sHa[tid][k] * sW3[k];
        const int gRow = wg * 16 + tid;
        out[BB + gRow] = acc;                       // Ei, flat [B*N]
        unsafeAtomicAdd(&out[gRow / NN], acc);      // Etot[b]
    }
}

// -------------------------------------------------------------------------
extern "C" void kernel_launch(void* const* d_in, const int* in_sizes, int n_in,
                              void* d_out, int out_size, void* d_ws, size_t ws_size,
                              hipStream_t stream)
{
    const float* image_dR   = (const float*)d_in[0];
    const int*   list_neigh = (const int*)d_in[1];
    const float* We0 = (const float*)d_in[2];
    const float* be0 = (const float*)d_in[3];
    const float* We1 = (const float*)d_in[4];
    const float* be1 = (const float*)d_in[5];
    const float* We2 = (const float*)d_in[6];
    const float* be2 = (const float*)d_in[7];
    const float* Wf0 = (const float*)d_in[8];
    const float* bf0 = (const float*)d_in[9];
    const float* Wf1 = (const float*)d_in[10];
    const float* bf1 = (const float*)d_in[11];
    const float* Wf2 = (const float*)d_in[12];
    const float* bf2 = (const float*)d_in[13];
    const float* Wf3 = (const float*)d_in[14];
    const float* bf3 = (const float*)d_in[15];
    float* out = (float*)d_out;

    float* dr = (float*)d_ws;   // B*N*1600 floats = 52.4 MB scratch

    k_embed_dr<<<BB * NN, AA, 0, stream>>>(image_dR, list_neigh,
                                           We0, be0, We1, be1, We2, be2, dr);
    k_zero_etot<<<1, 32, 0, stream>>>(out);
    k_fit<<<(BB * NN) / 16, 256, 0, stream>>>(dr, Wf0, bf0, Wf1, bf1,
                                              Wf2, bf2, Wf3, bf3, out);
}